// DenseLayer_32899449487452
// MI455X (gfx1250) — compile-verified
//
#include <hip/hip_runtime.h>

typedef float v2f __attribute__((ext_vector_type(2)));
typedef float v4f __attribute__((ext_vector_type(4)));
typedef float v8f __attribute__((ext_vector_type(8)));

#define ECOLS 256

// One wave processes a tile of 16 rows. Lane 'm' (and lane 'm+16') own row m:
//   lanes 0-15  : columns [0,128)   -> partials a0 (cols 0-63),  a1 (cols 64-127)
//   lanes 16-31 : columns [128,256) -> partials a0 (128-191),    a1 (192-255)
// This is exactly the V_WMMA_F32_16X16X4_F32 A-matrix layout (16x4, K = column
// quarter). One WMMA with B == ones reduces across the wave:
//   D[m][n] = sum_k A[m][k]  == full row-m dot product, replicated over n.
// D layout: VGPR r, lanes 0-15 -> row r; lanes 16-31 -> row r+8, so lane 0 and
// lane 16 each hold 8 consecutive per-row results -> two b128 stores.
__global__ __launch_bounds__(256) void dense_ratio_kernel(
    const float* __restrict__ x, const float* __restrict__ w,
    float* __restrict__ out, int n_rows)
{
    __shared__ float ws[ECOLS];
    ws[threadIdx.x] = w[threadIdx.x];          // blockDim.x == 256 == ECOLS
    __syncthreads();

    const int wave = threadIdx.x >> 5;         // 0..7
    const int lane = threadIdx.x & 31;
    const int half = lane >> 4;                // 0 or 1 (column half)
    const int m    = lane & 15;                // row within tile

    const int tile = blockIdx.x * 8 + wave;
    const int row0 = tile * 16;

    if (row0 + 16 <= n_rows) {
        // ---- full tile: WMMA-reduced path (wave-uniform branch, EXEC all 1s)
        const float* xr = x + (size_t)(row0 + m) * ECOLS + (size_t)half * 128;
        const float* wr = ws + half * 128;

        float a0s = 0.f, a0n = 0.f, a1s = 0.f, a1n = 0.f;
        #pragma unroll
        for (int c = 0; c < 128; c += 4) {
            // x is streamed exactly once: non-temporal b128 load
            v4f xv = __builtin_nontemporal_load((const v4f*)(xr + c));
            v4f wv = *(const v4f*)(wr + c);    // ds_read_b128, uniform broadcast
            float ps = 0.f, pn = 0.f;
            #pragma unroll
            for (int q = 0; q < 4; ++q) {
                float t = xv[q] * wv[q];       // t = x*w
                ps += t;                       // s   += x*w
                pn = fmaf(t, xv[q], pn);       // num += x*x*w
            }
            if (c < 64) { a0s += ps; a0n += pn; }
            else        { a1s += ps; a1n += pn; }
        }

        // Cross-lane reduction via f32 WMMA against a ones B-matrix.
        v2f As = {a0s, a1s};
        v2f An = {a0n, a1n};
        v2f B  = {1.0f, 1.0f};
        v8f Ds = {};
        v8f Dn = {};
        Ds = __builtin_amdgcn_wmma_f32_16x16x4_f32(
                 false, As, false, B, (short)0, Ds, false, false);
        Dn = __builtin_amdgcn_wmma_f32_16x16x4_f32(
                 false, An, false, B, (short)0, Dn, false, false);

        if (m == 0) {                          // lanes 0 and 16 write 8 rows each
            float o[8];
            #pragma unroll
            for (int r = 0; r < 8; ++r) {
                // Branchless: full-precision divide, then select 0 where s==0.
                float s = Ds[r];
                float q = Dn[r] / s;           // IEEE div; inf/nan if s==0 (discarded)
                o[r] = (s == 0.0f) ? 0.0f : q; // lowers to v_cndmask, not a branch
            }
            float* op = out + row0 + half * 8;
            *(v4f*)(op + 0) = (v4f){o[0], o[1], o[2], o[3]};
            *(v4f*)(op + 4) = (v4f){o[4], o[5], o[6], o[7]};
        }
    } else if (row0 < n_rows) {
        // ---- remainder tile (< 16 rows): simple scalar path, one row per lane
        const int row = row0 + lane;
        if (row < n_rows) {
            const float* xr = x + (size_t)row * ECOLS;
            float s = 0.f, nm = 0.f;
            #pragma unroll 8
            for (int c = 0; c < ECOLS; ++c) {
                float t = xr[c] * ws[c];
                s += t;
                nm = fmaf(t, xr[c], nm);
            }
            float q = nm / s;
            out[row] = (s == 0.0f) ? 0.0f : q;
        }
    }
}

extern "C" void kernel_launch(void* const* d_in, const int* in_sizes, int n_in,
                              void* d_out, int out_size, void* d_ws, size_t ws_size,
                              hipStream_t stream) {
    const float* x = (const float*)d_in[0];   // (N, 256) fp32
    const float* w = (const float*)d_in[1];   // (256, 1) fp32
    float* out = (float*)d_out;               // (N,) fp32

    const int n_rows = in_sizes[0] / ECOLS;
    const int tiles  = (n_rows + 15) / 16;
    const int blocks = (tiles + 7) / 8;       // 8 waves (tiles) per 256-thread block

    dense_ratio_kernel<<<dim3(blocks), dim3(256), 0, stream>>>(x, w, out, n_rows);
}